// LSTM_43800076485152
// MI455X (gfx1250) — compile-verified
//
#include <hip/hip_runtime.h>
#include <math.h>

// LSTM: T=512, B=64, I=H=1024, 4 gates.
//   Kernel 1 (parallel): Gx = X @ Wi^T + (bi+bh)  [M=32768,N=4096,K=1024]
//   Kernel 2 (x512, serial scan): g = Gx[t] + h_prev @ Wh^T, fused cell update.
// Both use one unified block shape: 256 threads = 8 waves covering a
// 32(M) x 64(N for k1) / 16(N per gate, x4 gates for k2) tile, with
// double-buffered GLOBAL_LOAD_ASYNC_TO_LDS_B128 staging (ASYNCcnt) feeding
// chained v_wmma_f32_16x16x4_f32.

typedef float v2f __attribute__((ext_vector_type(2)));
typedef float v8f __attribute__((ext_vector_type(8)));

#define TT 512
#define BB 64
#define II 1024
#define HH 1024
#define N4H 4096
// LDS row pitch in floats: 68*4 = 272 B -> every float4 seg 16B-aligned;
// 68 mod 64 = 4 -> 16 fragment lanes hit 16 distinct banks.
#define LDP 68

__device__ __forceinline__ float sigmoidf_(float x) {
  return 1.0f / (1.0f + __expf(-x));
}

// Issue (no wait) async copies for one K-chunk of 64 floats:
//   A: 32 rows x 64k cooperative across 256 threads (2 x b128 each)
//   W: this wave's 8 rows (half of its 16-row N-slice) (4 x b128 each)
__device__ __forceinline__ void stage_chunk_async(
    const float* __restrict__ A, int lda, int m0,
    const float* __restrict__ W, int ldw, int nbase,
    float* ldsAbuf, float* ldsBslice,
    int kb, int tid, int lane, int mt)
{
#pragma unroll
  for (int i = 0; i < 2; ++i) {
    int idx = tid + (i << 8);            // 0..511
    int row = idx >> 4, seg = idx & 15;  // 32 rows x 16 float4 segs
    const float* gp = A + (size_t)(m0 + row) * lda + kb + (seg << 2);
    unsigned lo = (unsigned)(size_t)(ldsAbuf + row * LDP + (seg << 2));
    asm volatile("global_load_async_to_lds_b128 %0, %1, off"
                 :: "v"(lo), "v"(gp) : "memory");
  }
#pragma unroll
  for (int i = 0; i < 4; ++i) {
    int idx = lane + (i << 5);           // 0..127
    int r = idx >> 4, seg = idx & 15;    // 8 rows x 16 segs
    int rr = (mt << 3) + r;              // row within the 16-row slice
    const float* gp = W + (size_t)(nbase + rr) * ldw + kb + (seg << 2);
    unsigned lo = (unsigned)(size_t)(ldsBslice + rr * LDP + (seg << 2));
    asm volatile("global_load_async_to_lds_b128 %0, %1, off"
                 :: "v"(lo), "v"(gp) : "memory");
  }
}

// acc += A[m0+mt*16 .. +15, :K] @ W[nbase .. nbase+15, :K]^T  (16x16 tile/wave)
// ldsA: float[2][32*LDP], ldsB: float[2][4][16*LDP]. Double-buffered chunks.
__device__ __forceinline__ void wave_gemm(
    v8f& acc,
    const float* __restrict__ A, int lda, int m0,
    const float* __restrict__ W, int ldw, int nbase,
    float* ldsA, float* ldsB,
    int K, int tid, int lane, int g, int mt)
{
  const int mh   = lane & 15;          // A row / W row within tile
  const int koff = (lane >> 4) << 1;   // 0 (lanes 0-15) / 2 (lanes 16-31)
  int buf = 0;

  stage_chunk_async(A, lda, m0, W, ldw, nbase,
                    ldsA, ldsB + g * (16 * LDP), 0, tid, lane, mt);
  asm volatile("s_wait_asynccnt 0" ::: "memory");
  __syncthreads();

  for (int kb = 0; kb < K; kb += 64) {
    if (kb + 64 < K) {  // prefetch next chunk into the other buffer
      int nb = buf ^ 1;
      stage_chunk_async(A, lda, m0, W, ldw, nbase,
                        ldsA + nb * (32 * LDP),
                        ldsB + (nb * 4 + g) * (16 * LDP),
                        kb + 64, tid, lane, mt);
    }
    const float* Ar = ldsA + buf * (32 * LDP) + (mt << 4) * LDP;
    const float* Bw = ldsB + (buf * 4 + g) * (16 * LDP);
    // 16 WMMA ops cover the K-chunk (K=4 each). Fragment layout per ISA:
    // VGPR0 = K at +0/+2 by lane half, VGPR1 = +1; n/m = lane&15.
#pragma unroll
    for (int kk = 0; kk < 64; kk += 4) {
      v2f a, b;
      a.x = Ar[mh * LDP + kk + koff];
      a.y = Ar[mh * LDP + kk + koff + 1];
      b.x = Bw[mh * LDP + kk + koff];
      b.y = Bw[mh * LDP + kk + koff + 1];
      acc = __builtin_amdgcn_wmma_f32_16x16x4_f32(
          false, a, false, b, (short)0, acc, false, false);
    }
    asm volatile("s_wait_asynccnt 0" ::: "memory");
    __syncthreads();
    buf ^= 1;
  }
}

// ---------------- Kernel 1: Gx = X @ Wi^T + (bi + bh) ----------------
// grid = (M/32, N/64), block = 256 (8 waves: n-slice = w&3, m-tile = w>>2)
__global__ void __launch_bounds__(256)
lstm_gx_gemm(const float* __restrict__ X,   // [T*B, I]
             const float* __restrict__ Wi,  // [4H, I]
             const float* __restrict__ bi,  // [4H]
             const float* __restrict__ bh,  // [4H]
             float* __restrict__ Gx)        // [T*B, 4H]
{
  __shared__ __align__(16) float ldsA[2][32 * LDP];
  __shared__ __align__(16) float ldsB[2][4][16 * LDP];

  const int tid  = threadIdx.x;
  const int lane = tid & 31;
  const int w    = tid >> 5;
  const int g    = w & 3;        // n-slice
  const int mt   = w >> 2;       // m half-tile
  const int m0   = blockIdx.x * 32;
  const int nb   = blockIdx.y * 64 + g * 16;

  v8f acc = {};
  wave_gemm(acc, X, II, m0, Wi, II, nb,
            &ldsA[0][0], &ldsB[0][0][0], II, tid, lane, g, mt);

  const int col   = nb + (lane & 15);
  const float bia = bi[col] + bh[col];
  const int rbase = (lane >> 4) << 3;  // D rows: +0 / +8 by lane half
#pragma unroll
  for (int r = 0; r < 8; ++r) {
    int row = m0 + (mt << 4) + rbase + r;
    Gx[(size_t)row * N4H + col] = acc[r] + bia;
  }
}

// ---------------- Kernel 2: one LSTM timestep ----------------
// grid = (B/32, H/16), block = 256 (8 waves: gate = w&3, m-tile = w>>2)
__global__ void __launch_bounds__(256)
lstm_step(const float* __restrict__ Gxt,    // [B, 4H] for this t (or null)
          const float* __restrict__ Xt,     // [B, I]  (fused fallback path)
          const float* __restrict__ Wi,     // [4H, I]
          const float* __restrict__ Wh,     // [4H, H]
          const float* __restrict__ bi,     // [4H]
          const float* __restrict__ bh,     // [4H]
          const float* __restrict__ hprev,  // [B, H] or null at t=0
          float* __restrict__ c,            // [B, H] cell state (ws)
          float* __restrict__ hout,         // [B, H] -> d_out + t*B*H
          int has_h, int use_gx)
{
  __shared__ __align__(16) float ldsA[2][32 * LDP];
  __shared__ __align__(16) float ldsB[2][4][16 * LDP];
  __shared__ float ldsG[4][32 * 16];

  const int tid  = threadIdx.x;
  const int lane = tid & 31;
  const int w    = tid >> 5;
  const int g    = w & 3;             // gate
  const int mt   = w >> 2;            // batch half-tile
  const int b0   = blockIdx.x * 32;   // batch tile
  const int h0   = blockIdx.y * 16;   // hidden tile
  const int nb   = g * HH + h0;       // row block in Wi/Wh, col block in Gx

  v8f acc = {};
  if (!use_gx)  // fallback: fuse the input projection into the scan
    wave_gemm(acc, Xt, II, b0, Wi, II, nb,
              &ldsA[0][0], &ldsB[0][0][0], II, tid, lane, g, mt);
  if (has_h)
    wave_gemm(acc, hprev, HH, b0, Wh, HH, nb,
              &ldsA[0][0], &ldsB[0][0][0], HH, tid, lane, g, mt);

  // Stage gate pre-activations into LDS (add Gx or bias).
  const int nh    = lane & 15;
  const int rbase = (lane >> 4) << 3;
  const float bia = use_gx ? 0.0f : (bi[nb + nh] + bh[nb + nh]);
#pragma unroll
  for (int r = 0; r < 8; ++r) {
    int row = (mt << 4) + rbase + r;  // batch row within 32-row tile
    float v = acc[r] + bia;
    if (use_gx) v += Gxt[(size_t)(b0 + row) * N4H + nb + nh];
    ldsG[g][row * 16 + nh] = v;
  }
  __syncthreads();

  // Fused cell update: 256 threads x 2 elements cover the 32x16 tile.
  for (int e = tid; e < 512; e += 256) {
    int bb = e >> 4, hh = e & 15;
    float f  = sigmoidf_(ldsG[0][e]);
    float i  = sigmoidf_(ldsG[1][e]);
    float gg = tanhf    (ldsG[2][e]);
    float o  = sigmoidf_(ldsG[3][e]);
    size_t idx = (size_t)(b0 + bb) * HH + h0 + hh;
    float cold = has_h ? c[idx] : 0.0f;
    float cn = f * cold + i * gg;
    c[idx]    = cn;
    hout[idx] = o * tanhf(cn);
  }
}

extern "C" void kernel_launch(void* const* d_in, const int* in_sizes, int n_in,
                              void* d_out, int out_size, void* d_ws, size_t ws_size,
                              hipStream_t stream) {
  (void)in_sizes; (void)n_in; (void)out_size;
  const float* X  = (const float*)d_in[0];  // [T,B,I]
  const float* Wi = (const float*)d_in[1];  // [4,H,I]
  const float* Wh = (const float*)d_in[2];  // [4,H,H]
  const float* bi = (const float*)d_in[3];  // [4,H]
  const float* bh = (const float*)d_in[4];  // [4,H]
  float* out = (float*)d_out;               // [T,B,H] ++ h_n[B,H] ++ c_n[B,H]

  // Workspace: c (256 KB) at offset 0, Gx (512 MB) at +1 MB.
  float* c_ws = (float*)d_ws;
  float* gx   = (float*)((char*)d_ws + (size_t)1024 * 1024);
  const size_t gx_bytes = sizeof(float) * (size_t)TT * BB * N4H;
  const int use_gx = (ws_size >= (size_t)1024 * 1024 + gx_bytes) ? 1 : 0;

  if (use_gx) {
    // M=32768, N=4096: grid (1024, 64); block = 32 rows x 64 cols.
    lstm_gx_gemm<<<dim3(1024, 64), 256, 0, stream>>>(X, Wi, bi, bh, gx);
  }

  const size_t stepBH = (size_t)BB * HH;
  for (int t = 0; t < TT; ++t) {
    const float* gxt   = use_gx ? gx + (size_t)t * BB * N4H : nullptr;
    const float* xt    = X + (size_t)t * BB * II;
    const float* hprev = (t > 0) ? out + (size_t)(t - 1) * stepBH : nullptr;
    lstm_step<<<dim3(BB / 32, HH / 16), 256, 0, stream>>>(
        gxt, xt, Wi, Wh, bi, bh, hprev, c_ws,
        out + (size_t)t * stepBH, (t > 0) ? 1 : 0, use_gx);
  }

  // h_n = outputs[T-1], c_n = cell state.
  hipMemcpyAsync(out + (size_t)TT * stepBH, out + (size_t)(TT - 1) * stepBH,
                 stepBH * sizeof(float), hipMemcpyDeviceToDevice, stream);
  hipMemcpyAsync(out + (size_t)TT * stepBH + stepBH, c_ws,
                 stepBH * sizeof(float), hipMemcpyDeviceToDevice, stream);
}